// Kernel_Conv_54932631715975
// MI455X (gfx1250) — compile-verified
//
#include <hip/hip_runtime.h>
#include <hip/hip_bf16.h>
#include <stdint.h>

typedef __attribute__((ext_vector_type(2))) float v2f;
typedef __attribute__((ext_vector_type(8))) float v8f;

typedef __attribute__((address_space(1))) int gint;   // global int (AS1)
typedef __attribute__((address_space(3))) int lint;   // LDS int (AS3)

#define Bsz 2
#define CH  16
#define Hh  128
#define Ww  128
#define HW  (Hh*Ww)
#define KD  19
#define PADK 9
#define NP  (KD*KD)   // 361
#define CS  18        // padded channel stride (even -> 8B align, conflict-free)

// ---------------------------------------------------------------------------
// Stage 1 & 3: 3x3 conv (16->16 ch) + activation, via V_WMMA_F32_16X16X4_F32.
// One block per (batch, row): 8 waves = 8 w-tiles of 16 pixels. LDS layouts
// put both K-values of every WMMA fragment contiguously so each A/B fragment
// is a single aligned ds_load_b64 feeding the WMMA operand pair directly.
// ACT: 0 = exact GELU, 1 = sigmoid.
// ---------------------------------------------------------------------------
template <int ACT>
__global__ __launch_bounds__(256) void conv3x3_wmma(
    const float* __restrict__ x,     // [B,16,128,128]
    const float* __restrict__ wgt,   // [16,16,3,3]
    const float* __restrict__ bias,  // [16]
    float* __restrict__ out)         // [B,16,128,128]
{
    __shared__ float w_t[9 * CH * CS];    // [tap][oc][c], 2592 floats
    __shared__ float b_s[CH];
    __shared__ float xs[3 * 130 * CS];    // [r][col][c],  7020 floats

    const int bid = blockIdx.x;          // B*H = 256 blocks
    const int h   = bid & (Hh - 1);
    const int tb  = bid >> 7;

    // transpose weights [o][c][tap] -> [tap][o][c] (coalesced global reads)
    for (int i = threadIdx.x; i < CH * CH * 9; i += 256) {
        const int tap = i % 9;
        const int c   = (i / 9) % CH;
        const int o   = i / 144;
        w_t[(tap * CH + o) * CS + c] = wgt[i];
    }
    if (threadIdx.x < CH) b_s[threadIdx.x] = bias[threadIdx.x];

    // halo strip rows h-1..h+1, cols -1..128 (zero-padded), [r][col][c]
    const float* xb = x + (size_t)tb * CH * HW;
    for (int i = threadIdx.x; i < CH * 3 * 130; i += 256) {
        const int col = i % 130;          // coalesced in col
        const int r   = (i / 130) % 3;
        const int c   = i / 390;
        const int gh = h + r - 1;
        const int gw = col - 1;
        float v = 0.0f;
        if (gh >= 0 && gh < Hh && gw >= 0 && gw < Ww)
            v = xb[((size_t)c * Hh + gh) * Ww + gw];
        xs[(r * 130 + col) * CS + c] = v;
    }
    __syncthreads();

    const int wave = threadIdx.x >> 5;   // w-tile 0..7
    const int lane = threadIdx.x & 31;
    const int half = lane >> 4;          // 0: K0/K1, M=r ; 1: K2/K3, M=r+8
    const int l    = lane & 15;          // oc for A, pixel for B/D
    const int w0   = wave * 16;

    v8f acc = {};

    #pragma unroll
    for (int kh = 0; kh < 3; ++kh) {
        #pragma unroll
        for (int kw = 0; kw < 3; ++kw) {
            const int col  = w0 + l + kw;            // 0..129, always in-strip
            const int tap  = kh * 3 + kw;
            const float* ap = &w_t[(tap * CH + l) * CS];
            const float* bp = &xs[(kh * 130 + col) * CS];
            #pragma unroll
            for (int cg = 0; cg < 4; ++cg) {
                const int cb = cg * 4 + half * 2;    // channel pair, even
                const v2f a = *(const v2f*)(ap + cb);  // ds_load_b64
                const v2f b = *(const v2f*)(bp + cb);  // ds_load_b64
                acc = __builtin_amdgcn_wmma_f32_16x16x4_f32(
                    false, a, false, b, (short)0, acc, false, false);
            }
        }
    }

    // D layout: lane half {0,1}, N = l, VGPR r -> M = r + half*8 (out channel)
    float* ob = out + (size_t)tb * CH * HW + (size_t)h * Ww + (w0 + l);
    #pragma unroll
    for (int r = 0; r < 8; ++r) {
        const int M = r + half * 8;
        float v = acc[r] + b_s[M];
        if (ACT == 0) {
            v = 0.5f * v * (1.0f + erff(v * 0.70710678118654752f));  // exact GELU
        } else {
            v = 1.0f / (1.0f + expf(-v));                            // sigmoid
        }
        ob[(size_t)M * HW] = v;
    }
}

// ---------------------------------------------------------------------------
// Stage 2: per-pixel spatially-varying 19x19 conv, shared across channels.
// Memory-bound on kern (47 MB): each kern value is loaded once (coalesced)
// and amortized over 16 channels in registers; the y halo is staged in LDS
// via gfx1250 async global->LDS copies.
// ---------------------------------------------------------------------------
__global__ __launch_bounds__(256) void dynconv19(
    const float* __restrict__ y,     // [B,16,128,128]
    const float* __restrict__ kern,  // [B,361,128,128]
    float* __restrict__ out)         // [B,16,128,128]
{
    extern __shared__ __align__(16) float ys[];  // 34*34*16 floats = 73,984 B

    const int bx  = blockIdx.x;          // 128 blocks = B * 8 * 8
    const int txb = bx & 7;
    const int tyb = (bx >> 3) & 7;
    const int b   = bx >> 6;
    const int w0  = txb * 16;
    const int h0  = tyb * 16;

    const float* yb = y + (size_t)b * CH * HW;

    // cooperative halo load: iterate [c][row][col] for coalesced global reads,
    // store transposed into LDS [row][col][c]
    const int NH = 34 * 34 * CH;
    for (int i = threadIdx.x; i < NH; i += 256) {
        const int col = i % 34;
        const int row = (i / 34) % 34;
        const int c   = i / (34 * 34);
        const int gh = h0 + row - PADK;
        const int gw = w0 + col - PADK;
        float* lp = &ys[(row * 34 + col) * CH + c];
        if (gh >= 0 && gh < Hh && gw >= 0 && gw < Ww) {
            const float* gp = &yb[((size_t)c * Hh + gh) * Ww + gw];
#if __has_builtin(__builtin_amdgcn_global_load_async_to_lds_b32)
            // ASYNCcnt-tracked direct global->LDS copy (no VGPR round trip).
            __builtin_amdgcn_global_load_async_to_lds_b32(
                (gint*)(uintptr_t)gp, (lint*)(uintptr_t)lp, 0, 0);
#else
            *lp = *gp;
#endif
        } else {
            *lp = 0.0f;
        }
    }
#if __has_builtin(__builtin_amdgcn_global_load_async_to_lds_b32)
#if __has_builtin(__builtin_amdgcn_s_wait_asynccnt)
    __builtin_amdgcn_s_wait_asynccnt(0);
#else
    asm volatile("s_wait_asynccnt 0x0" ::: "memory");
#endif
#endif
    __syncthreads();

    const int tx = threadIdx.x & 15;
    const int ty = threadIdx.x >> 4;
    const int hh = h0 + ty;
    const int ww = w0 + tx;
    const float* kp = kern + ((size_t)b * NP * Hh + hh) * Ww + ww;

    float acc[CH];
    #pragma unroll
    for (int c = 0; c < CH; ++c) acc[c] = 0.0f;

    #pragma unroll 1
    for (int dh = 0; dh < KD; ++dh) {
        if (dh < KD - 1) {
            // pull the next tap-row of the HBM-bound kern stream into cache
            __builtin_prefetch(kp + (size_t)((dh + 1) * KD) * HW, 0, 0);
        }
        #pragma unroll
        for (int dw = 0; dw < KD; ++dw) {
            const float kv = kp[(size_t)(dh * KD + dw) * HW];
            const float* yr = &ys[((ty + dh) * 34 + (tx + dw)) * CH];
            const float4 v0 = *(const float4*)(yr + 0);
            const float4 v1 = *(const float4*)(yr + 4);
            const float4 v2 = *(const float4*)(yr + 8);
            const float4 v3 = *(const float4*)(yr + 12);
            acc[0]  += v0.x * kv; acc[1]  += v0.y * kv;
            acc[2]  += v0.z * kv; acc[3]  += v0.w * kv;
            acc[4]  += v1.x * kv; acc[5]  += v1.y * kv;
            acc[6]  += v1.z * kv; acc[7]  += v1.w * kv;
            acc[8]  += v2.x * kv; acc[9]  += v2.y * kv;
            acc[10] += v2.z * kv; acc[11] += v2.w * kv;
            acc[12] += v3.x * kv; acc[13] += v3.y * kv;
            acc[14] += v3.z * kv; acc[15] += v3.w * kv;
        }
    }

    float* ob = out + (size_t)b * CH * HW + (size_t)hh * Ww + ww;
    #pragma unroll
    for (int c = 0; c < CH; ++c) ob[(size_t)c * HW] = acc[c];
}

// ---------------------------------------------------------------------------
extern "C" void kernel_launch(void* const* d_in, const int* in_sizes, int n_in,
                              void* d_out, int out_size, void* d_ws, size_t ws_size,
                              hipStream_t stream) {
    const float* input = (const float*)d_in[0];  // [2,16,128,128]
    const float* kern  = (const float*)d_in[1];  // [2,361,128,128]
    const float* w1    = (const float*)d_in[2];  // [16,16,3,3]
    const float* b1    = (const float*)d_in[3];  // [16]
    const float* w2    = (const float*)d_in[4];  // [16,16,3,3]
    const float* b2    = (const float*)d_in[5];  // [16]
    float* outp = (float*)d_out;

    const size_t n = (size_t)Bsz * CH * HW;      // 524288
    float* y1 = (float*)d_ws;                    // conv1+GELU result
    float* y2 = y1 + n;                          // dynamic-conv result

    // grid = B*H = 256 blocks; 8 waves per block = 8 w-tiles per row
    conv3x3_wmma<0><<<256, 256, 0, stream>>>(input, w1, b1, y1);

    const int lds = 34 * 34 * CH * sizeof(float);  // 73,984 B (<320KB/WGP)
    dynconv19<<<128, 256, lds, stream>>>(y1, kern, y2);

    conv3x3_wmma<1><<<256, 256, 0, stream>>>(y2, w2, b2, outp);
}